// GC_NET_63788854280228
// MI455X (gfx1250) — compile-verified
//
#include <hip/hip_runtime.h>

#define N_NODES   50000
#define N_EDGES   800000
#define F_DIM     64
#define N_GRAPHS  128
#define N_CLASSES 10

typedef float v2f __attribute__((ext_vector_type(2)));
typedef float v8f __attribute__((ext_vector_type(8)));

// ---------------------------------------------------------------------------
// 0) init: deg = 1.0 (self-loop), agg = 0, pool sums/counts = 0
// ---------------------------------------------------------------------------
__global__ void init_kernel(float* __restrict__ deg, float* __restrict__ agg,
                            float* __restrict__ psum, float* __restrict__ pcnt) {
    int t = blockIdx.x * blockDim.x + threadIdx.x;
    if (t < N_NODES) deg[t] = 1.0f;                    // self-loop contribution
    if (t < N_NODES * F_DIM) agg[t] = 0.0f;
    if (t < N_GRAPHS * F_DIM) psum[t] = 0.0f;
    if (t < N_GRAPHS) pcnt[t] = 0.0f;
}

// ---------------------------------------------------------------------------
// 1) in-degree over edges (dst side)
// ---------------------------------------------------------------------------
__global__ void degree_kernel(const int* __restrict__ dst, float* __restrict__ deg) {
    int e = blockIdx.x * blockDim.x + threadIdx.x;
    if (e < N_EDGES) atomicAdd(&deg[dst[e]], 1.0f);
}

// ---------------------------------------------------------------------------
// 2) dis = deg^{-1/2}, in place (deg >= 1 always)
// ---------------------------------------------------------------------------
__global__ void rsqrt_kernel(float* __restrict__ deg) {
    int i = blockIdx.x * blockDim.x + threadIdx.x;
    if (i < N_NODES) deg[i] = rsqrtf(deg[i]);
}

// ---------------------------------------------------------------------------
// 3) h = x @ W_conv  via V_WMMA_F32_16X16X4_F32 (wave32).
//    One wave -> 16x64 output tile: 4 accumulators (v8f), 16 K-steps.
//    A 16x4 f32 layout: lane L holds A[L&15][khalf], A[L&15][khalf+1],
//      khalf = (L>=16) ? 2 : 0.
//    B 4x16 f32 layout (row striped across lanes per VGPR):
//      VGPR j, lane L -> B[khalf + j][L&15].
//    C/D 16x16 f32: VGPR v, lane L -> D[v + 8*(L>>4)][L&15].
// ---------------------------------------------------------------------------
__global__ __launch_bounds__(256) void gemm_wmma_f32(
    const float* __restrict__ x, const float* __restrict__ W,
    float* __restrict__ h, int nrows)
{
    __shared__ float Ws[F_DIM * F_DIM];                 // 16 KB of the 320 KB/WGP LDS
    for (int i = threadIdx.x; i < F_DIM * F_DIM; i += 256) Ws[i] = W[i];
    __syncthreads();

    int wave = blockIdx.x * 8 + (threadIdx.x >> 5);
    int lane = threadIdx.x & 31;
    int row0 = wave * 16;
    if (row0 >= nrows) return;                          // 50000 % 16 == 0 -> no ragged tile

    int m     = lane & 15;
    int khalf = (lane >> 4) << 1;                       // 0 or 2
    const float* xr = x + (size_t)(row0 + m) * F_DIM;

    v8f c0 = {}, c1 = {}, c2 = {}, c3 = {};
#pragma unroll
    for (int kk = 0; kk < 16; ++kk) {
        int kb = kk * 4 + khalf;
        v2f a;  a.x = xr[kb];  a.y = xr[kb + 1];
        v2f b0, b1, b2, b3;
        b0.x = Ws[(kb + 0) * F_DIM + m +  0]; b0.y = Ws[(kb + 1) * F_DIM + m +  0];
        b1.x = Ws[(kb + 0) * F_DIM + m + 16]; b1.y = Ws[(kb + 1) * F_DIM + m + 16];
        b2.x = Ws[(kb + 0) * F_DIM + m + 32]; b2.y = Ws[(kb + 1) * F_DIM + m + 32];
        b3.x = Ws[(kb + 0) * F_DIM + m + 48]; b3.y = Ws[(kb + 1) * F_DIM + m + 48];
        c0 = __builtin_amdgcn_wmma_f32_16x16x4_f32(false, a, false, b0, (short)0, c0, false, false);
        c1 = __builtin_amdgcn_wmma_f32_16x16x4_f32(false, a, false, b1, (short)0, c1, false, false);
        c2 = __builtin_amdgcn_wmma_f32_16x16x4_f32(false, a, false, b2, (short)0, c2, false, false);
        c3 = __builtin_amdgcn_wmma_f32_16x16x4_f32(false, a, false, b3, (short)0, c3, false, false);
    }

    int rbase = row0 + ((lane >> 4) << 3);
#pragma unroll
    for (int v = 0; v < 8; ++v) {
        float* hr = h + (size_t)(rbase + v) * F_DIM + m;
        hr[ 0] = c0[v];  hr[16] = c1[v];  hr[32] = c2[v];  hr[48] = c3[v];
    }
}

// ---------------------------------------------------------------------------
// 4) edge aggregation: agg[dst] += h[src] * dis[src]*dis[dst]
//    64 consecutive threads per edge -> coalesced 256B gather + scatter.
//    h (12.8 MB) is L2-resident (192 MB L2) so atomics and gathers stay on-chip.
// ---------------------------------------------------------------------------
__global__ void aggregate_kernel(const int* __restrict__ src, const int* __restrict__ dst,
                                 const float* __restrict__ dis, const float* __restrict__ h,
                                 float* __restrict__ agg) {
    long long t = (long long)blockIdx.x * blockDim.x + threadIdx.x;
    int e = (int)(t >> 6);
    if (e >= N_EDGES) return;
    int f = (int)(t & 63);
    int s = src[e], d = dst[e];
    float norm = dis[s] * dis[d];
    atomicAdd(&agg[d * F_DIM + f], h[s * F_DIM + f] * norm);
}

// ---------------------------------------------------------------------------
// 5) fold self-loop term, ReLU, mean-pool accumulation per graph
// ---------------------------------------------------------------------------
__global__ void pool_kernel(const float* __restrict__ agg, const float* __restrict__ h,
                            const float* __restrict__ dis, const int* __restrict__ batch,
                            float* __restrict__ psum, float* __restrict__ pcnt) {
    long long t = (long long)blockIdx.x * blockDim.x + threadIdx.x;
    int i = (int)(t >> 6);
    if (i >= N_NODES) return;
    int f = (int)(t & 63);
    float dii = dis[i];
    float v = agg[i * F_DIM + f] + h[i * F_DIM + f] * dii * dii;   // self-loop msg
    v = fmaxf(v, 0.0f);                                            // relu
    int b = batch[i];
    atomicAdd(&psum[b * F_DIM + f], v);
    if (f == 0) atomicAdd(&pcnt[b], 1.0f);
}

// ---------------------------------------------------------------------------
// 6) g = psum / max(cnt,1);  logits = g @ W_lin   (tiny: one block)
// ---------------------------------------------------------------------------
__global__ __launch_bounds__(128) void final_kernel(
    const float* __restrict__ psum, const float* __restrict__ pcnt,
    const float* __restrict__ Wl, float* __restrict__ out) {
    __shared__ float Ws[F_DIM * N_CLASSES];
    for (int i = threadIdx.x; i < F_DIM * N_CLASSES; i += 128) Ws[i] = Wl[i];
    __syncthreads();
    int b = threadIdx.x;
    if (b >= N_GRAPHS) return;
    float inv = 1.0f / fmaxf(pcnt[b], 1.0f);
    float g[F_DIM];
#pragma unroll
    for (int f = 0; f < F_DIM; ++f) g[f] = psum[b * F_DIM + f] * inv;
#pragma unroll
    for (int k = 0; k < N_CLASSES; ++k) {
        float acc = 0.0f;
#pragma unroll
        for (int f = 0; f < F_DIM; ++f) acc = fmaf(g[f], Ws[f * N_CLASSES + k], acc);
        out[b * N_CLASSES + k] = acc;
    }
}

// ---------------------------------------------------------------------------
extern "C" void kernel_launch(void* const* d_in, const int* in_sizes, int n_in,
                              void* d_out, int out_size, void* d_ws, size_t ws_size,
                              hipStream_t stream) {
    const float* x     = (const float*)d_in[0];   // [N, 64]
    const int*   eidx  = (const int*)  d_in[1];   // [2, E]
    const int*   batch = (const int*)  d_in[2];   // [N]
    const float* Wc    = (const float*)d_in[3];   // [64, 64]
    const float* Wl    = (const float*)d_in[4];   // [64, 10]
    float*       out   = (float*)d_out;           // [128, 10]

    const int* src = eidx;                        // edge_index[0]
    const int* dst = eidx + N_EDGES;              // edge_index[1]

    // workspace carve-up (all offsets 256B-aligned)
    char*  ws   = (char*)d_ws;
    float* deg  = (float*)(ws + 0);               // N floats (becomes dis)
    float* h    = (float*)(ws +   262144);        // N*64 floats = 12.8 MB
    float* agg  = (float*)(ws + 13062144);        // N*64 floats = 12.8 MB
    float* psum = (float*)(ws + 25862144);        // 128*64 floats
    float* pcnt = (float*)(ws + 25894912);        // 128 floats

    // 0) init (covers largest buffer: N*64 = 3.2M elems)
    init_kernel<<<(N_NODES * F_DIM + 255) / 256, 256, 0, stream>>>(deg, agg, psum, pcnt);
    // 1) degrees
    degree_kernel<<<(N_EDGES + 255) / 256, 256, 0, stream>>>(dst, deg);
    // 2) deg^{-1/2}
    rsqrt_kernel<<<(N_NODES + 255) / 256, 256, 0, stream>>>(deg);
    // 3) h = x @ W_conv  (3125 waves of 16 rows; 8 waves/block)
    gemm_wmma_f32<<<(3125 + 7) / 8, 256, 0, stream>>>(x, Wc, h, N_NODES);
    // 4) edge aggregation (64 threads per edge)
    aggregate_kernel<<<(int)(((long long)N_EDGES * 64 + 255) / 256), 256, 0, stream>>>(
        src, dst, deg, h, agg);
    // 5) relu + mean-pool accumulate
    pool_kernel<<<(int)(((long long)N_NODES * 64 + 255) / 256), 256, 0, stream>>>(
        agg, h, deg, batch, psum, pcnt);
    // 6) logits
    final_kernel<<<1, 128, 0, stream>>>(psum, pcnt, Wl, out);
}